// SpeakerLoss_14920716386625
// MI455X (gfx1250) — compile-verified
//
#include <hip/hip_runtime.h>
#include <hip/hip_bf16.h>

typedef float v2f __attribute__((ext_vector_type(2)));
typedef float v8f __attribute__((ext_vector_type(8)));

#define BB 16
#define SS_ 4
#define TT 1000
#define DD 256
#define NN 8192
#define TCHUNKS 8
#define TPC 125           // frames per chunk
#define EMB_STRIDE 258    // 258 % 64 == 2 -> conflict-free column access

__device__ __forceinline__ float waveRedSum(float v) {
#pragma unroll
    for (int off = 16; off > 0; off >>= 1) v += __shfl_xor(v, off, 32);
    return v;
}
__device__ __forceinline__ float waveRedMin(float v) {
#pragma unroll
    for (int off = 16; off > 0; off >>= 1) v = fminf(v, __shfl_xor(v, off, 32));
    return v;
}

// ---------------------------------------------------------------------------
// Kernel 1: partial weighted average of per-frame-normalized speaker vectors.
// grid = (B*S, TCHUNKS), block = 256 (8 waves). Wave handles frames strided 8.
// Writes e_part[c][bs][256] and act_part[c][bs] (no atomics -> deterministic).
// ---------------------------------------------------------------------------
__global__ __launch_bounds__(256) void k_wavg_part(
    const float* __restrict__ spks, const float* __restrict__ ts,
    const int* __restrict__ sigmas, const int* __restrict__ ilens,
    float* __restrict__ e_part, float* __restrict__ act_part) {
    const int bs = blockIdx.x;          // 0..63
    const int c  = blockIdx.y;          // 0..7
    const int b  = bs >> 2;
    const int tid = threadIdx.x, wv = tid >> 5, lane = tid & 31;
    const int sig  = sigmas[bs];
    const int ilen = ilens[b];
    const float* base = spks + (long)bs * TT * DD;

    float acc[8] = {0.f,0.f,0.f,0.f,0.f,0.f,0.f,0.f};
    float actAcc = 0.f;
    const int tEnd = (c + 1) * TPC;
    for (int t = c * TPC + wv; t < tEnd; t += 8) {
        float wt_raw = ts[((long)b * TT + t) * SS_ + sig];
        bool valid = (t < ilen);
        const float* fr = base + (long)t * DD + lane * 8;
        float v[8]; float ssq = 0.f;
#pragma unroll
        for (int j = 0; j < 8; ++j) { v[j] = fr[j]; ssq += v[j] * v[j]; }
        ssq = waveRedSum(ssq);                      // ||frame||^2 (wave-wide)
        float w  = valid ? wt_raw : 0.f;
        float rn = rsqrtf(ssq) * w;
#pragma unroll
        for (int j = 0; j < 8; ++j) acc[j] += v[j] * rn;
        actAcc += valid ? rintf(wt_raw) : 0.f;      // round-half-even like jnp
    }

    __shared__ float red[8][DD];
    __shared__ float ared[8];
#pragma unroll
    for (int j = 0; j < 8; ++j) red[wv][lane * 8 + j] = acc[j];
    if (lane == 0) ared[wv] = actAcc;
    __syncthreads();
    float sum = 0.f;
#pragma unroll
    for (int w2 = 0; w2 < 8; ++w2) sum += red[w2][tid];
    e_part[((long)c * (BB * SS_) + bs) * DD + tid] = sum;
    if (tid == 0) {
        float a = 0.f;
        for (int w2 = 0; w2 < 8; ++w2) a += ared[w2];
        act_part[c * (BB * SS_) + bs] = a;
    }
}

// ---------------------------------------------------------------------------
// Kernel 2: combine chunks, L2-normalize -> e[bs][256]; active flag.
// grid = B*S, block = 256.
// ---------------------------------------------------------------------------
__global__ __launch_bounds__(256) void k_finalize(
    const float* __restrict__ e_part, const float* __restrict__ act_part,
    float* __restrict__ e_ws, int* __restrict__ active_ws) {
    const int bs = blockIdx.x, tid = threadIdx.x;
    const int wv = tid >> 5, lane = tid & 31;
    float w = 0.f;
#pragma unroll
    for (int c = 0; c < TCHUNKS; ++c)
        w += e_part[((long)c * (BB * SS_) + bs) * DD + tid];
    float sq = waveRedSum(w * w);
    __shared__ float ls[8];
    __shared__ float rnS;
    if (lane == 0) ls[wv] = sq;
    __syncthreads();
    if (tid == 0) {
        float t = 0.f;
        for (int i = 0; i < 8; ++i) t += ls[i];
        rnS = rsqrtf(t);
        float a = 0.f;
        for (int c = 0; c < TCHUNKS; ++c) a += act_part[c * (BB * SS_) + bs];
        active_ws[bs] = (a > 0.f) ? 1 : 0;
    }
    __syncthreads();
    e_ws[bs * DD + tid] = w * rnS;
}

// ---------------------------------------------------------------------------
// Kernel 3 (WMMA): d[b,s,n] = max(alpha,eps)*max(2-2*e.emb_n,0)+beta.
// grid = (N/32, B), block = 64 (2 waves). Each wave owns a 16-candidate tile.
// A-tile (16x256, rows 4..15 zero) staged once in LDS -> inner loop is two
// aligned ds_load_b64 + one V_WMMA_F32_16X16X4_F32 per K-step, no exec masks.
// ---------------------------------------------------------------------------
__global__ __launch_bounds__(64) void k_dist(
    const float* __restrict__ table, const int* __restrict__ ns,
    const float* __restrict__ e_ws, const float* __restrict__ alphaP,
    const float* __restrict__ betaP, float* __restrict__ dmat) {
    const int b = blockIdx.y;
    const int nbase = blockIdx.x * 32;
    const int tid = threadIdx.x;
    const int wv = tid >> 5, lane = tid & 31;

    __shared__ float embS[32][EMB_STRIDE];
    __shared__ float eS[16][EMB_STRIDE];   // full A tile, rows 4..15 zeroed

    // stage zero-padded A tile: rows 0..3 = e[b], rows 4..15 = 0
    for (int i = tid; i < 16 * DD; i += 64) {
        const int row = i >> 8, col = i & 255;
        eS[row][col] = (row < SS_) ? e_ws[b * (SS_ * DD) + row * DD + col] : 0.f;
    }

    // gather + normalize 32 candidate rows (wave handles every other row)
    for (int r = wv; r < 32; r += 2) {
        int row = ns[b * NN + nbase + r];
        const float* src = table + (long)row * DD + lane * 8;
        float v[8]; float ssq = 0.f;
#pragma unroll
        for (int j = 0; j < 8; ++j) { v[j] = src[j]; ssq += v[j] * v[j]; }
        ssq = waveRedSum(ssq);
        float rn = rsqrtf(ssq);
#pragma unroll
        for (int j = 0; j < 8; ++j) embS[r][lane * 8 + j] = v[j] * rn;
    }
    __syncthreads();

    // WMMA operand indices per ISA 16x16x4 f32 layout:
    //  A: lane = M (mod 16), VGPR0/1 = K {0,1} (lanes 0-15) / {2,3} (lanes 16-31)
    //  B: lane = N (mod 16), same K split
    const int m     = lane & 15;            // A row (M) / B-D column (N)
    const int nloc  = (wv << 4) + m;        // candidate within block
    const int khalf = (lane >> 4) << 1;     // K sub-offset: lanes 16-31 -> +2

    v8f acc = {};
#pragma unroll 4
    for (int kk = 0; kk < 64; ++kk) {
        const int k = kk * 4 + khalf;       // even -> 8B-aligned LDS reads
        v2f a  = *(const v2f*)&eS[m][k];
        v2f bf = *(const v2f*)&embS[nloc][k];
        acc = __builtin_amdgcn_wmma_f32_16x16x4_f32(
            false, a, false, bf, (short)0, acc, false, false);
    }

    const float alpha = fmaxf(alphaP[0], 2.2204460492503131e-16f);
    const float beta  = betaP[0];
    if (lane < 16) {   // D rows M=0..3 live in VGPRs 0..3 of lanes 0..15
#pragma unroll
        for (int j = 0; j < SS_; ++j) {
            float cosv  = acc[j];
            float dist2 = fmaxf(2.f - 2.f * cosv, 0.f);
            dmat[((long)(b * SS_ + j)) * NN + nbase + nloc] = alpha * dist2 + beta;
        }
    }
}

// ---------------------------------------------------------------------------
// Kernel 4: loss_bs = active * (d[idx] + logsumexp(-d)).  grid=B*S, block=256.
// ---------------------------------------------------------------------------
__global__ __launch_bounds__(256) void k_loss(
    const float* __restrict__ dmat, const int* __restrict__ ss,
    const int* __restrict__ sigmas, const int* __restrict__ active_ws,
    float* __restrict__ loss_bs) {
    const int bs = blockIdx.x, b = bs >> 2;
    const int tid = threadIdx.x, wv = tid >> 5, lane = tid & 31;
    const float* row = dmat + (long)bs * NN;

    float mn = 3.402823466e38f;
    for (int i = tid; i < NN; i += 256) mn = fminf(mn, row[i]);
    mn = waveRedMin(mn);
    __shared__ float ls[8];
    __shared__ float mnS;
    if (lane == 0) ls[wv] = mn;
    __syncthreads();
    if (tid == 0) {
        float t = ls[0];
        for (int i = 1; i < 8; ++i) t = fminf(t, ls[i]);
        mnS = t;
    }
    __syncthreads();
    const float dmin = mnS;

    float se = 0.f;
    for (int i = tid; i < NN; i += 256) se += expf(dmin - row[i]);
    se = waveRedSum(se);
    __syncthreads();
    if (lane == 0) ls[wv] = se;
    __syncthreads();
    if (tid == 0) {
        float t = 0.f;
        for (int i = 0; i < 8; ++i) t += ls[i];
        int idx = ss[b * SS_ + sigmas[bs]];        // ss[b, sigma[b,s]]
        float loss = row[idx] - dmin + logf(t);    // d[idx] + LSE(-d)
        loss_bs[bs] = active_ws[bs] ? loss : 0.f;
    }
}

// ---------------------------------------------------------------------------
// Kernel 5: mean over 64 entries. grid=1, block=64.
// ---------------------------------------------------------------------------
__global__ __launch_bounds__(64) void k_mean(const float* __restrict__ loss_bs,
                                             float* __restrict__ out) {
    float v = loss_bs[threadIdx.x];
    v = waveRedSum(v);
    __shared__ float t2[2];
    if ((threadIdx.x & 31) == 0) t2[threadIdx.x >> 5] = v;
    __syncthreads();
    if (threadIdx.x == 0) out[0] = (t2[0] + t2[1]) * (1.0f / (BB * SS_));
}

extern "C" void kernel_launch(void* const* d_in, const int* in_sizes, int n_in,
                              void* d_out, int out_size, void* d_ws, size_t ws_size,
                              hipStream_t stream) {
    const float* spks   = (const float*)d_in[0];
    // d_in[1] = ys (unused by the math)
    const float* ts     = (const float*)d_in[2];
    const int*   ss     = (const int*)d_in[3];
    const int*   sigmas = (const int*)d_in[4];
    const int*   ns     = (const int*)d_in[5];
    const int*   ilens  = (const int*)d_in[6];
    const float* table  = (const float*)d_in[7];
    const float* alphaP = (const float*)d_in[8];
    const float* betaP  = (const float*)d_in[9];
    float* out = (float*)d_out;

    char* ws = (char*)d_ws;
    float* e_part    = (float*)(ws + 0);          // 8*64*256 f = 512 KB
    float* act_part  = (float*)(ws + 524288);     // 512 f
    float* e_ws      = (float*)(ws + 526336);     // 64*256 f = 64 KB
    int*   active_ws = (int*)  (ws + 591872);     // 64 i
    float* loss_bs   = (float*)(ws + 592128);     // 64 f
    float* dmat      = (float*)(ws + 592384);     // 64*8192 f = 2 MB

    k_wavg_part<<<dim3(BB * SS_, TCHUNKS), 256, 0, stream>>>(
        spks, ts, sigmas, ilens, e_part, act_part);
    k_finalize<<<BB * SS_, 256, 0, stream>>>(e_part, act_part, e_ws, active_ws);
    k_dist<<<dim3(NN / 32, BB), 64, 0, stream>>>(
        table, ns, e_ws, alphaP, betaP, dmat);
    k_loss<<<BB * SS_, 256, 0, stream>>>(dmat, ss, sigmas, active_ws, loss_bs);
    k_mean<<<1, 64, 0, stream>>>(loss_bs, out);
}